// PenaltyHuberLoss_80341658239294
// MI455X (gfx1250) — compile-verified
//
#include <hip/hip_runtime.h>

// ext-vector types: real LLVM vectors so __builtin_nontemporal_load works and
// loads lower to global_load_b128 with TH=NT.
typedef float f32x4 __attribute__((ext_vector_type(4)));
typedef int   i32x4 __attribute__((ext_vector_type(4)));
typedef float v2f   __attribute__((ext_vector_type(2)));
typedef float v8f   __attribute__((ext_vector_type(8)));

#define BLOCKS 2048   // must stay a multiple of 64 (finisher consumes 64/iter)
#define TPB    256

// weighted Huber term, DELTA = 0.5, penalty LUT {1,5,4,2} packed as nibbles.
__device__ __forceinline__ float huber_w(float p, int t) {
    int cls = t < 0 ? 0 : (t > 3 ? 3 : t);
    float w  = (float)((0x2451u >> (cls << 2)) & 0xFu);  // 1,5,4,2
    float tf = (float)t;
    float d  = p - tf;
    float ad = fabsf(d);
    float h  = (ad < 0.5f) ? (0.5f * d * d) : (0.5f * (ad - 0.25f));
    return h * w;
}

__global__ __launch_bounds__(TPB)
void phl_partial_kernel(const float* __restrict__ pred,
                        const int*   __restrict__ ytrue,
                        float*       __restrict__ partials,
                        int n) {
    const int tid      = blockIdx.x * TPB + threadIdx.x;
    const int nthreads = gridDim.x * TPB;
    const int nv4      = n >> 2;

    float acc = 0.0f;
    // vectorized streaming pass: b128 loads, non-temporal (single-use data)
    for (int i = tid; i < nv4; i += nthreads) {
        f32x4 p = __builtin_nontemporal_load(((const f32x4*)pred)  + i);
        i32x4 t = __builtin_nontemporal_load(((const i32x4*)ytrue) + i);
        acc += huber_w(p.x, t.x);
        acc += huber_w(p.y, t.y);
        acc += huber_w(p.z, t.z);
        acc += huber_w(p.w, t.w);
    }
    // scalar tail (N % 4, empty for N = 2^24 but keep it correct)
    for (int i = (nv4 << 2) + tid; i < n; i += nthreads)
        acc += huber_w(pred[i], ytrue[i]);

    // wave32 butterfly reduction
    #pragma unroll
    for (int m = 16; m >= 1; m >>= 1)
        acc += __shfl_xor(acc, m, 32);

    __shared__ float lds[TPB / 32];
    const int lane = threadIdx.x & 31;
    const int wave = threadIdx.x >> 5;
    if (lane == 0) lds[wave] = acc;
    __syncthreads();
    if (wave == 0) {
        float v = (lane < (TPB / 32)) ? lds[lane] : 0.0f;
        #pragma unroll
        for (int m = 4; m >= 1; m >>= 1)
            v += __shfl_xor(v, m, 32);
        if (lane == 0) partials[blockIdx.x] = v;
    }
}

// Single-wave finisher. Reduces BLOCKS floats with V_WMMA_F32_16X16X4_F32:
// A = all-ones 16x4  =>  D[m][n] = sum_k B[k][n] (+C), so each WMMA folds a
// 64-value chunk into 16 per-column accumulators. Because A is all-ones the
// B layout is irrelevant (every element lands in exactly one column sum), so
// each lane loads a CONTIGUOUS pair -> one coalesced global_load_b64, no
// bounds checks, no EXEC manipulation (WMMA requires EXEC all-ones anyway).
__global__ __launch_bounds__(32)
void phl_final_kernel(const float* __restrict__ partials,
                      float*       __restrict__ out,
                      float inv_n) {
    const int lane = threadIdx.x;            // 0..31, full wave
    v2f a; a.x = 1.0f; a.y = 1.0f;           // all-ones A matrix (16x4 f32)
    v8f c = {};
    #pragma unroll 4
    for (int base = 0; base < BLOCKS; base += 64) {
        v2f b = *(const v2f*)(partials + base + (lane << 1));  // b64, coalesced
        // 8 args: (neg_a, A, neg_b, B, c_mod, C, reuse_a, reuse_b)
        c = __builtin_amdgcn_wmma_f32_16x16x4_f32(
                false, a, false, b, (short)0, c, false, false);
    }
    // D VGPR0: lanes 0-15 hold column sums (row M=0); lanes 16-31 duplicate (M=8).
    float v = c[0];
    #pragma unroll
    for (int m = 8; m >= 1; m >>= 1)   // reduce within lanes 0..15 only
        v += __shfl_xor(v, m, 32);
    if (lane == 0) out[0] = v * inv_n;
}

extern "C" void kernel_launch(void* const* d_in, const int* in_sizes, int n_in,
                              void* d_out, int out_size, void* d_ws, size_t ws_size,
                              hipStream_t stream) {
    const float* pred  = (const float*)d_in[0];
    const int*   ytrue = (const int*)d_in[1];
    float*       out   = (float*)d_out;
    float*       parts = (float*)d_ws;          // BLOCKS * 4 = 8 KiB scratch
    const int    n     = in_sizes[0];

    phl_partial_kernel<<<BLOCKS, TPB, 0, stream>>>(pred, ytrue, parts, n);
    phl_final_kernel<<<1, 32, 0, stream>>>(parts, out, 1.0f / (float)n);
}